// GAT_51153060495581
// MI455X (gfx1250) — compile-verified
//
#include <hip/hip_runtime.h>
#include <math.h>

#define NN 2048           // feature / node dim
#define NB 8              // batch
#define NBINS 65536

typedef __attribute__((ext_vector_type(2))) float v2f;
typedef __attribute__((ext_vector_type(8))) float v8f;

// ---------------------------------------------------------------------------
// K1: row softmax of adj -> a   (one block per row)
// ---------------------------------------------------------------------------
__global__ __launch_bounds__(256) void k_row_softmax(const float* __restrict__ adj,
                                                     float* __restrict__ a) {
  __shared__ float red[256];
  const int i = blockIdx.x;
  const int t = threadIdx.x;
  const float* row = adj + (size_t)i * NN;

  float mymax = -INFINITY;
  for (int j = t; j < NN; j += 256) mymax = fmaxf(mymax, row[j]);
  red[t] = mymax; __syncthreads();
  for (int s = 128; s > 0; s >>= 1) { if (t < s) red[t] = fmaxf(red[t], red[t + s]); __syncthreads(); }
  const float m = red[0];
  __syncthreads();

  float sum = 0.f;
  for (int j = t; j < NN; j += 256) sum += __expf(row[j] - m);
  red[t] = sum; __syncthreads();
  for (int s = 128; s > 0; s >>= 1) { if (t < s) red[t] += red[t + s]; __syncthreads(); }
  const float inv = 1.f / red[0];

  float* arow = a + (size_t)i * NN;
  for (int j = t; j < NN; j += 256) arow[j] = __expf(row[j] - m) * inv;
}

// ---------------------------------------------------------------------------
// K2..K5: exact lower median of all N*N softmax values via 2-level 16-bit
// radix select on the float bit pattern (all values are positive => monotonic)
// ---------------------------------------------------------------------------
__global__ __launch_bounds__(256) void k_hist_hi(const float* __restrict__ a,
                                                 unsigned* __restrict__ hist) {
  const size_t total = (size_t)NN * NN;
  const size_t stride = (size_t)gridDim.x * blockDim.x;
  for (size_t p = (size_t)blockIdx.x * blockDim.x + threadIdx.x; p < total; p += stride)
    atomicAdd(&hist[__float_as_uint(a[p]) >> 16], 1u);
}

__global__ void k_scan_hi(const unsigned* __restrict__ hist, unsigned* __restrict__ meta) {
  if (threadIdx.x == 0 && blockIdx.x == 0) {
    const unsigned k = ((unsigned)NN * (unsigned)NN - 1u) / 2u;  // torch lower median
    unsigned cum = 0;
    for (unsigned bin = 0; bin < NBINS; ++bin) {
      const unsigned c = hist[bin];
      if (cum + c > k) { meta[0] = bin; meta[1] = cum; break; }
      cum += c;
    }
  }
}

__global__ __launch_bounds__(256) void k_hist_lo(const float* __restrict__ a,
                                                 const unsigned* __restrict__ meta,
                                                 unsigned* __restrict__ hist2) {
  const unsigned sel = meta[0];
  const size_t total = (size_t)NN * NN;
  const size_t stride = (size_t)gridDim.x * blockDim.x;
  for (size_t p = (size_t)blockIdx.x * blockDim.x + threadIdx.x; p < total; p += stride) {
    const unsigned bits = __float_as_uint(a[p]);
    if ((bits >> 16) == sel) atomicAdd(&hist2[bits & 0xFFFFu], 1u);
  }
}

__global__ void k_scan_lo(const unsigned* __restrict__ hist2, unsigned* __restrict__ meta) {
  if (threadIdx.x == 0 && blockIdx.x == 0) {
    const unsigned k = ((unsigned)NN * (unsigned)NN - 1u) / 2u;
    const unsigned kk = k - meta[1];
    unsigned cum = 0;
    for (unsigned low = 0; low < NBINS; ++low) {
      const unsigned c = hist2[low];
      if (cum + c > kk) { meta[2] = (meta[0] << 16) | low; break; }
      cum += c;
    }
  }
}

// ---------------------------------------------------------------------------
// K6: Y[b, n] = sum_j x[b, j] * W[n, j]   (the only true GEMM after the
// rank-1 factorization) using V_WMMA_F32_16X16X4_F32.
// One wave (32 threads) per 16-wide n-tile; M=16 (batch zero-padded 8->16 in
// LDS so the hot loop has NO predication: pure ds_load_b64 + v_wmma).
// A 16x4 layout: lanes 0-15 hold K={k,k+1} of row M=lane; lanes 16-31 hold
// K={k+2,k+3} of row M=lane-16.  B 4x16 mirrored (lane = N column).
// Two accumulators break the WMMA C-dependency chain.
// ---------------------------------------------------------------------------
__global__ __launch_bounds__(32) void k_gemm_y(const float* __restrict__ x,
                                               const float* __restrict__ W,
                                               float* __restrict__ Y) {
  __shared__ float Wt[16][132];     // [n-offset][k-offset], padded vs bank conflicts
  __shared__ float Xt[16][132];     // [m-row][k-offset], rows 8..15 stay zero
  const int nb   = blockIdx.x * 16;
  const int lane = threadIdx.x;
  const int half = lane >> 4;       // 0 or 1
  const int l16  = lane & 15;

  // zero the pad rows once (they are never rewritten)
  for (int idx = lane; idx < 8 * 128; idx += 32)
    Xt[8 + (idx >> 7)][idx & 127] = 0.f;

  v8f acc0 = {}, acc1 = {};
  for (int k0 = 0; k0 < NN; k0 += 128) {
    // stage x[0..7][k0..k0+127] and W[nb..nb+15][k0..k0+127]; branch-free, coalesced
    for (int idx = lane; idx < 8 * 128; idx += 32) {
      const int r = idx >> 7, c = idx & 127;
      Xt[r][c] = x[r * NN + (k0 + c)];
    }
    for (int idx = lane; idx < 16 * 128; idx += 32) {
      const int r = idx >> 7, c = idx & 127;
      Wt[r][c] = W[(size_t)(nb + r) * NN + (k0 + c)];
    }
    __syncthreads();
    #pragma unroll 4
    for (int kk = 0; kk < 128; kk += 8) {
      const int ka0 = kk + half * 2;
      v2f a0 = { Xt[l16][ka0], Xt[l16][ka0 + 1] };
      v2f b0 = { Wt[l16][ka0], Wt[l16][ka0 + 1] };
      acc0 = __builtin_amdgcn_wmma_f32_16x16x4_f32(false, a0, false, b0,
                                                   (short)0, acc0, false, false);
      const int ka1 = kk + 4 + half * 2;
      v2f a1 = { Xt[l16][ka1], Xt[l16][ka1 + 1] };
      v2f b1 = { Wt[l16][ka1], Wt[l16][ka1 + 1] };
      acc1 = __builtin_amdgcn_wmma_f32_16x16x4_f32(false, a1, false, b1,
                                                   (short)0, acc1, false, false);
    }
    __syncthreads();
  }
  // C/D 16x16 f32 layout: VGPR r -> M=r (lanes 0-15) / M=r+8 (lanes 16-31)
  for (int r = 0; r < 8; ++r) {
    const int m = r + half * 8;
    if (m < NB) Y[m * NN + nb + l16] = acc0[r] + acc1[r];
  }
}

// ---------------------------------------------------------------------------
// K7: per-row softmax stats of lrelu(x_bi * Y[b,:] + bias): rowM (max), rowZ (sum)
// ---------------------------------------------------------------------------
__global__ __launch_bounds__(256) void k_row_stats(const float* __restrict__ x,
                                                   const float* __restrict__ Y,
                                                   const float* __restrict__ bias,
                                                   float* __restrict__ rowM,
                                                   float* __restrict__ rowZ) {
  __shared__ float lr[NN];
  __shared__ float red[256];
  const int i = blockIdx.x;
  const int b = blockIdx.y;
  const int t = threadIdx.x;
  const float xbi = x[b * NN + i];

  float mymax = -INFINITY;
  for (int j = t; j < NN; j += 256) {
    float v = fmaf(xbi, Y[b * NN + j], bias[j]);
    v = (v > 0.f) ? v : 0.01f * v;           // leaky relu
    lr[j] = v;
    mymax = fmaxf(mymax, v);
  }
  red[t] = mymax; __syncthreads();
  for (int s = 128; s > 0; s >>= 1) { if (t < s) red[t] = fmaxf(red[t], red[t + s]); __syncthreads(); }
  const float m = red[0];
  __syncthreads();

  float sum = 0.f;
  for (int j = t; j < NN; j += 256) sum += __expf(lr[j] - m);
  red[t] = sum; __syncthreads();
  for (int s = 128; s > 0; s >>= 1) { if (t < s) red[t] += red[t + s]; __syncthreads(); }
  if (t == 0) { rowM[b * NN + i] = m; rowZ[b * NN + i] = red[0]; }
}

// ---------------------------------------------------------------------------
// K8: out[b,j] = sum_i (x_bi / Z_bi) * [a_ij > med] * exp(lrelu(x_bi*y_bj+b_j) - m_bi)
// grid (N/256, B, 8); each block owns a 256-wide j-tile and a 256-long i-range,
// accumulates locally, then one fp32 atomicAdd per thread.
// ---------------------------------------------------------------------------
__global__ __launch_bounds__(256) void k_aggregate(const float* __restrict__ x,
                                                   const float* __restrict__ Y,
                                                   const float* __restrict__ bias,
                                                   const float* __restrict__ a,
                                                   const unsigned* __restrict__ meta,
                                                   const float* __restrict__ rowM,
                                                   const float* __restrict__ rowZ,
                                                   float* __restrict__ out) {
  const int j  = blockIdx.x * 256 + threadIdx.x;
  const int b  = blockIdx.y;
  const int iBase = blockIdx.z * (NN / 8);     // 256 i's per z-slice
  const float med = __uint_as_float(meta[2]);
  const float ybj = Y[b * NN + j];
  const float bj  = bias[j];

  __shared__ float sx[128], sm[128], sc[128];
  float acc = 0.f;
  for (int i0 = iBase; i0 < iBase + 256; i0 += 128) {
    if (threadIdx.x < 128) {
      const int i = i0 + threadIdx.x;
      const float xv = x[b * NN + i];
      sx[threadIdx.x] = xv;
      sm[threadIdx.x] = rowM[b * NN + i];
      sc[threadIdx.x] = xv / rowZ[b * NN + i];
    }
    __syncthreads();
    for (int ii = 0; ii < 128; ++ii) {
      const int i = i0 + ii;
      if ((ii & 15) == 0)   // strided column walk of `a`: hint the prefetcher
        __builtin_prefetch(&a[(size_t)(i + 16) * NN + j], 0, 1);
      const float aij = a[(size_t)i * NN + j];
      float t = fmaf(sx[ii], ybj, bj);
      t = (t > 0.f) ? t : 0.01f * t;
      acc += (aij > med) ? sc[ii] * __expf(t - sm[ii]) : 0.f;
    }
    __syncthreads();
  }
  atomicAdd(&out[b * NN + j], acc);
}

// ---------------------------------------------------------------------------
extern "C" void kernel_launch(void* const* d_in, const int* in_sizes, int n_in,
                              void* d_out, int out_size, void* d_ws, size_t ws_size,
                              hipStream_t stream) {
  const float* x    = (const float*)d_in[0];
  const float* adj  = (const float*)d_in[1];
  const float* W    = (const float*)d_in[2];
  const float* bias = (const float*)d_in[3];
  float* out = (float*)d_out;

  char* ws = (char*)d_ws;
  size_t off = 0;
  float*    a     = (float*)(ws + off);    off += (size_t)NN * NN * sizeof(float);   // 16 MB
  float*    Y     = (float*)(ws + off);    off += (size_t)NB * NN * sizeof(float);
  float*    rowM  = (float*)(ws + off);    off += (size_t)NB * NN * sizeof(float);
  float*    rowZ  = (float*)(ws + off);    off += (size_t)NB * NN * sizeof(float);
  unsigned* hist1 = (unsigned*)(ws + off); off += (size_t)NBINS * sizeof(unsigned);
  unsigned* hist2 = (unsigned*)(ws + off); off += (size_t)NBINS * sizeof(unsigned);
  unsigned* meta  = (unsigned*)(ws + off); off += 16;

  hipMemsetAsync(hist1, 0, 2 * (size_t)NBINS * sizeof(unsigned), stream);  // hist1+hist2
  hipMemsetAsync(out, 0, (size_t)out_size * sizeof(float), stream);

  k_row_softmax<<<NN, 256, 0, stream>>>(adj, a);
  k_hist_hi   <<<1024, 256, 0, stream>>>(a, hist1);
  k_scan_hi   <<<1, 32, 0, stream>>>(hist1, meta);
  k_hist_lo   <<<1024, 256, 0, stream>>>(a, meta, hist2);
  k_scan_lo   <<<1, 32, 0, stream>>>(hist2, meta);
  k_gemm_y    <<<NN / 16, 32, 0, stream>>>(x, W, Y);
  k_row_stats <<<dim3(NN, NB), 256, 0, stream>>>(x, Y, bias, rowM, rowZ);
  k_aggregate <<<dim3(NN / 256, NB, 8), 256, 0, stream>>>(x, Y, bias, a, meta, rowM, rowZ, out);
}